// DeformableMNIST_48137993453667
// MI455X (gfx1250) — compile-verified
//
#include <hip/hip_runtime.h>
#include <hip/hip_bf16.h>
#include <math.h>

typedef float v2f __attribute__((ext_vector_type(2)));
typedef float v8f __attribute__((ext_vector_type(8)));

#define DEV static __device__ __forceinline__

// Bilinear sample with zero outside bounds (matches reference: clip-gather * valid mask)
DEV float bil1(const float* __restrict__ img, int H, int W, float py, float px) {
  float y0f = floorf(py), x0f = floorf(px);
  float wy = py - y0f, wx = px - x0f;
  int y0 = (int)y0f, x0 = (int)x0f;
  int y1 = y0 + 1, x1 = x0 + 1;
  float v00 = (y0 >= 0 && y0 < H && x0 >= 0 && x0 < W) ? img[y0 * W + x0] : 0.f;
  float v01 = (y0 >= 0 && y0 < H && x1 >= 0 && x1 < W) ? img[y0 * W + x1] : 0.f;
  float v10 = (y1 >= 0 && y1 < H && x0 >= 0 && x0 < W) ? img[y1 * W + x0] : 0.f;
  float v11 = (y1 >= 0 && y1 < H && x1 >= 0 && x1 < W) ? img[y1 * W + x1] : 0.f;
  return (1.f - wy) * ((1.f - wx) * v00 + wx * v01) + wy * ((1.f - wx) * v10 + wx * v11);
}

// ---------------------------------------------------------------------------
// Stage 1 fused: offset conv (1->18) + deformable sampling + 1->32 conv.
// One thread per (b, y, x) of the 28x28 map.  Writes h1 (256,32,28,28).
// ---------------------------------------------------------------------------
__global__ __launch_bounds__(256) void stage1_fused(
    const float* __restrict__ x,       // (256,1,28,28)
    const float* __restrict__ off_w1,  // (18,1,3,3) -> [18][9]
    const float* __restrict__ off_b1,  // (18)
    const float* __restrict__ w1,      // (32,1,3,3) -> [32][9]
    const float* __restrict__ b1,      // (32)
    float* __restrict__ h1)            // (256,32,28,28)
{
  const int HW = 28 * 28;
  int t = blockIdx.x * blockDim.x + threadIdx.x;
  if (t >= 256 * HW) return;
  int b = t / HW;
  int yx = t - b * HW;
  int y = yx / 28, xx = yx - y * 28;
  const float* xb = x + b * HW;

  // 3x3 zero-padded patch of the single input channel
  float patch[9];
#pragma unroll
  for (int k = 0; k < 9; k++) {
    int py = y + k / 3 - 1, px = xx + k % 3 - 1;
    patch[k] = (py >= 0 && py < 28 && px >= 0 && px < 28) ? xb[py * 28 + px] : 0.f;
  }

  float samp[9];
#pragma unroll
  for (int k = 0; k < 9; k++) {
    float dy = off_b1[2 * k];
    float dx = off_b1[2 * k + 1];
#pragma unroll
    for (int j = 0; j < 9; j++) {
      dy = fmaf(off_w1[(2 * k) * 9 + j], patch[j], dy);
      dx = fmaf(off_w1[(2 * k + 1) * 9 + j], patch[j], dx);
    }
    float py = (float)y + (float)(k / 3 - 1) + dy;
    float px = (float)xx + (float)(k % 3 - 1) + dx;
    samp[k] = bil1(xb, 28, 28, py, px);
  }

#pragma unroll
  for (int o = 0; o < 32; o++) {
    float acc = b1[o];
#pragma unroll
    for (int k = 0; k < 9; k++) acc = fmaf(w1[o * 9 + k], samp[k], acc);
    h1[(b * 32 + o) * HW + yx] = acc;
  }
}

// ---------------------------------------------------------------------------
// ReLU + 2x2 max pool (relu commutes with max).
// ---------------------------------------------------------------------------
__global__ __launch_bounds__(256) void relu_pool(
    const float* __restrict__ in, float* __restrict__ out,
    int C, int Hin, int Win)
{
  int Ho = Hin >> 1, Wo = Win >> 1;
  int total = 256 * C * Ho * Wo;
  int t = blockIdx.x * blockDim.x + threadIdx.x;
  if (t >= total) return;
  int wo = t % Wo;
  int r = t / Wo;
  int ho = r % Ho; r /= Ho;
  int c = r % C;
  int b = r / C;
  const float* p = in + ((b * C + c) * Hin + ho * 2) * Win + wo * 2;
  float m = fmaxf(fmaxf(p[0], p[1]), fmaxf(p[Win], p[Win + 1]));
  out[t] = fmaxf(m, 0.f);
}

// ---------------------------------------------------------------------------
// Offset conv for stage 2: (256,32,14,14) -> (256,18,14,14), zero pad.
// ---------------------------------------------------------------------------
__global__ __launch_bounds__(256) void conv_off2(
    const float* __restrict__ in,   // (256,32,14,14)
    const float* __restrict__ w,    // (18,32,3,3) -> [18][32][9]
    const float* __restrict__ bias, // (18)
    float* __restrict__ out)        // (256,18,14,14)
{
  const int HW = 196;
  int total = 256 * 18 * HW;
  int t = blockIdx.x * blockDim.x + threadIdx.x;
  if (t >= total) return;
  int yx = t % HW;
  int r = t / HW;
  int o = r % 18;
  int b = r / 18;
  int y = yx / 14, x = yx - y * 14;
  float acc = bias[o];
  const float* ib = in + b * 32 * HW;
  const float* wo = w + o * 288;
  for (int c = 0; c < 32; c++) {
    const float* ic = ib + c * HW;
    const float* wc = wo + c * 9;
#pragma unroll
    for (int k = 0; k < 9; k++) {
      int py = y + k / 3 - 1, px = x + k % 3 - 1;
      if (py >= 0 && py < 14 && px >= 0 && px < 14)
        acc = fmaf(wc[k], ic[py * 14 + px], acc);
    }
  }
  out[t] = acc;
}

// ---------------------------------------------------------------------------
// Stage 2 deformable conv as WMMA GEMM:
//   D[m, o] = sum_{k<288} S[m, k] * w2t[k, o],  m = b*196 + y*14 + x
// M = 50176, N = 64, K = 288 (= 32 channels x 9 taps, index c*9+kk).
// 4 waves / block, 16 positions per wave, V_WMMA_F32_16X16X4_F32.
// ---------------------------------------------------------------------------
#define SROW 292                 // padded S row stride in floats (bank-friendly)
#define LDSW_FLOATS (288 * 64)   // transposed weights
#define LDSS_FLOATS (4 * 16 * SROW)
#define SMEM_BYTES ((LDSW_FLOATS + LDSS_FLOATS) * 4)

__global__ __launch_bounds__(128) void stage2_deform_wmma(
    const float* __restrict__ p1,   // (256,32,14,14)
    const float* __restrict__ off2, // (256,18,14,14)
    const float* __restrict__ w2,   // (64,32,3,3) -> [64][288]
    const float* __restrict__ b2,   // (64)
    float* __restrict__ h2)         // (256,64,14,14)
{
  extern __shared__ float smem[];
  float* lds_w = smem;                 // [288][64] : lds_w[k*64+n] = w2[n][k]
  float* lds_s = smem + LDSW_FLOATS;   // [4][16][SROW]

  const int HW = 196;
  const int lane = threadIdx.x & 31;
  const int wave = threadIdx.x >> 5;

  // Stage transposed weights (whole block cooperates).
  for (int i = threadIdx.x; i < 288 * 64; i += blockDim.x) {
    int k = i >> 6, n = i & 63;
    lds_w[i] = w2[n * 288 + k];
  }

  // Bilinear sampling: this wave fills its own 16 x 288 S tile.
  int mbase = (blockIdx.x * 4 + wave) * 16;
  float* S = lds_s + wave * 16 * SROW;
  for (int task = lane; task < 16 * 9; task += 32) {
    int m_local = task / 9, k = task - m_local * 9;
    int m = mbase + m_local;
    int b = m / HW;
    int yx = m - b * HW;
    int y = yx / 14, x = yx - y * 14;

    const float* ob = off2 + b * 18 * HW;
    float dy = ob[(2 * k) * HW + yx];
    float dx = ob[(2 * k + 1) * HW + yx];
    float py = (float)y + (float)(k / 3 - 1) + dy;
    float px = (float)x + (float)(k % 3 - 1) + dx;

    float y0f = floorf(py), x0f = floorf(px);
    float wy = py - y0f, wx = px - x0f;
    int y0 = (int)y0f, x0 = (int)x0f;
    int y1 = y0 + 1, x1 = x0 + 1;
    int y0c = min(max(y0, 0), 13), x0c = min(max(x0, 0), 13);
    int y1c = min(max(y1, 0), 13), x1c = min(max(x1, 0), 13);
    float f00 = (1.f - wy) * (1.f - wx) * ((y0 >= 0 && y0 < 14 && x0 >= 0 && x0 < 14) ? 1.f : 0.f);
    float f01 = (1.f - wy) * wx         * ((y0 >= 0 && y0 < 14 && x1 >= 0 && x1 < 14) ? 1.f : 0.f);
    float f10 = wy * (1.f - wx)         * ((y1 >= 0 && y1 < 14 && x0 >= 0 && x0 < 14) ? 1.f : 0.f);
    float f11 = wy * wx                 * ((y1 >= 0 && y1 < 14 && x1 >= 0 && x1 < 14) ? 1.f : 0.f);
    int i00 = y0c * 14 + x0c, i01 = y0c * 14 + x1c;
    int i10 = y1c * 14 + x0c, i11 = y1c * 14 + x1c;

    const float* pb = p1 + b * 32 * HW;
    float* Srow = S + m_local * SROW + k;   // feature index c*9 + k
#pragma unroll 4
    for (int c = 0; c < 32; c++) {
      const float* pc = pb + c * HW;
      float v = f00 * pc[i00] + f01 * pc[i01] + f10 * pc[i10] + f11 * pc[i11];
      Srow[c * 9] = v;
    }
  }
  __syncthreads();

  // GEMM: 72 chunks of K=4, four 16x16 N tiles.
  v8f acc0 = {0.f, 0.f, 0.f, 0.f, 0.f, 0.f, 0.f, 0.f};
  v8f acc1 = acc0, acc2 = acc0, acc3 = acc0;

  const float* Sme = S + (lane & 15) * SROW;
  const int koff = (lane < 16) ? 0 : 2;

  for (int k4 = 0; k4 < 72; k4++) {
    int kk = k4 * 4 + koff;
    v2f a;
    a.x = Sme[kk];
    a.y = Sme[kk + 1];
    const float* Wk0 = lds_w + kk * 64 + (lane & 15);
    const float* Wk1 = Wk0 + 64;
    v2f bf0; bf0.x = Wk0[0];  bf0.y = Wk1[0];
    v2f bf1; bf1.x = Wk0[16]; bf1.y = Wk1[16];
    v2f bf2; bf2.x = Wk0[32]; bf2.y = Wk1[32];
    v2f bf3; bf3.x = Wk0[48]; bf3.y = Wk1[48];
    acc0 = __builtin_amdgcn_wmma_f32_16x16x4_f32(false, a, false, bf0, (short)0, acc0, false, false);
    acc1 = __builtin_amdgcn_wmma_f32_16x16x4_f32(false, a, false, bf1, (short)0, acc1, false, false);
    acc2 = __builtin_amdgcn_wmma_f32_16x16x4_f32(false, a, false, bf2, (short)0, acc2, false, false);
    acc3 = __builtin_amdgcn_wmma_f32_16x16x4_f32(false, a, false, bf3, (short)0, acc3, false, false);
  }

  // Store D: VGPR r -> M = r (lanes 0-15) / r+8 (lanes 16-31), N = lane&15.
  int n = lane & 15;
  int mofs = (lane < 16) ? 0 : 8;
#pragma unroll
  for (int r = 0; r < 8; r++) {
    int m = mbase + r + mofs;
    int b = m / HW;
    int yx = m - b * HW;
    float* ob = h2 + b * 64 * HW + yx;
    ob[(n +  0) * HW] = acc0[r] + b2[n +  0];
    ob[(n + 16) * HW] = acc1[r] + b2[n + 16];
    ob[(n + 32) * HW] = acc2[r] + b2[n + 32];
    ob[(n + 48) * HW] = acc3[r] + b2[n + 48];
  }
}

// ---------------------------------------------------------------------------
// Final FC: (256,3136) x (10,3136)^T + bias
// ---------------------------------------------------------------------------
__global__ __launch_bounds__(256) void fc_kernel(
    const float* __restrict__ p2,   // (256, 3136)
    const float* __restrict__ fw,   // (10, 3136)
    const float* __restrict__ fb,   // (10)
    float* __restrict__ out)        // (256, 10)
{
  int t = blockIdx.x * blockDim.x + threadIdx.x;
  if (t >= 256 * 10) return;
  int b = t / 10, j = t - b * 10;
  const float* row = p2 + b * 3136;
  const float* wr = fw + j * 3136;
  float acc = fb[j];
  for (int i = 0; i < 3136; i++) acc = fmaf(row[i], wr[i], acc);
  out[t] = acc;
}

// ---------------------------------------------------------------------------
extern "C" void kernel_launch(void* const* d_in, const int* in_sizes, int n_in,
                              void* d_out, int out_size, void* d_ws, size_t ws_size,
                              hipStream_t stream) {
  (void)in_sizes; (void)n_in; (void)out_size; (void)ws_size;
  const float* x      = (const float*)d_in[0];
  const float* off_w1 = (const float*)d_in[1];
  const float* off_b1 = (const float*)d_in[2];
  const float* w1     = (const float*)d_in[3];
  const float* b1     = (const float*)d_in[4];
  const float* off_w2 = (const float*)d_in[5];
  const float* off_b2 = (const float*)d_in[6];
  const float* w2     = (const float*)d_in[7];
  const float* b2     = (const float*)d_in[8];
  const float* fc_w   = (const float*)d_in[9];
  const float* fc_b   = (const float*)d_in[10];
  float* out = (float*)d_out;

  // Workspace layout (floats). h2/p2 reuse the dead h1 region.
  float* ws   = (float*)d_ws;
  float* h1   = ws;                        // 256*32*784 = 6,422,528
  float* p1   = h1 + 6422528;              // 256*32*196 = 1,605,632
  float* off2 = p1 + 1605632;              // 256*18*196 =   903,168
  float* h2   = ws;                        // 256*64*196 = 3,211,264 (reuse h1)
  float* p2   = ws + 3211264;              // 256*64*49  =   802,816 (after h2)

  // 1) fused stage-1 deformable conv: 200704 px / 256 = 784 blocks
  stage1_fused<<<784, 256, 0, stream>>>(x, off_w1, off_b1, w1, b1, h1);

  // 2) relu + pool 28->14 : 1,605,632 / 256 = 6272 blocks
  relu_pool<<<6272, 256, 0, stream>>>(h1, p1, 32, 28, 28);

  // 3) offset conv stage 2 : 903,168 / 256 = 3528 blocks
  conv_off2<<<3528, 256, 0, stream>>>(p1, off_w2, off_b2, off2);

  // 4) deformable conv stage 2 via WMMA: 50176 positions / 64 per block = 784
  stage2_deform_wmma<<<784, 128, SMEM_BYTES, stream>>>(p1, off2, w2, b2, h2);

  // 5) relu + pool 14->7 : 802,816 / 256 = 3136 blocks
  relu_pool<<<3136, 256, 0, stream>>>(h2, p2, 64, 14, 14);

  // 6) fully connected : 2560 threads
  fc_kernel<<<10, 256, 0, stream>>>(p2, fc_w, fc_b, out);
}